// Max_loss_62251255988863
// MI455X (gfx1250) — compile-verified
//
#include <hip/hip_runtime.h>
#include <hip/hip_bf16.h>

typedef __attribute__((ext_vector_type(2))) float v2f;
typedef __attribute__((ext_vector_type(8))) float v8f;

#define W_DIM     512
#define N_ELEM    (64 * 512 * 512)          // 16,777,216
#define N_CHUNK   (N_ELEM / 4)              // 4,194,304 float4 chunks
#define BLOCKS    2048
#define THREADS   256
#define ITERS     (N_CHUNK / (BLOCKS * THREADS))   // == 8, exact
#define FIN_ITERS (BLOCKS / THREADS)               // == 8, exact
#define SIG_W     30.0f
#define CLOSE_MIN 0.05f

// Full 32-lane sum via two V_WMMA_F32_16X16X4_F32 ops.
// WMMA #1: A = (v, 0) per lane -> A[m][0]=v_m, A[m][2]=v_{m+16}; B = ones.
//   D[m][n] = v_m + v_{m+16}. Lane L<16 holds D[0..7][L]; lane L>=16 holds
//   D[8..15][L-16]. Summing the 8 D regs gives half-sums t_lo / t_hi.
// WMMA #2 on t: D2[m][n] = t_lo + t_hi = total, broadcast to every lane.
// Requires EXEC all-1s (callers guarantee no divergence here).
__device__ __forceinline__ float wave_reduce_wmma(float v) {
    v2f a;  a[0] = v;    a[1] = 0.0f;
    v2f b;  b[0] = 1.0f; b[1] = 1.0f;          // all-ones B: layout-independent
    v8f c = {0.f, 0.f, 0.f, 0.f, 0.f, 0.f, 0.f, 0.f};

    v8f d = __builtin_amdgcn_wmma_f32_16x16x4_f32(
        false, a, false, b, (short)0, c, false, false);
    float t = ((d[0] + d[1]) + (d[2] + d[3])) + ((d[4] + d[5]) + (d[6] + d[7]));

    v2f a2; a2[0] = t; a2[1] = 0.0f;
    v8f d2 = __builtin_amdgcn_wmma_f32_16x16x4_f32(
        false, a2, false, b, (short)0, c, false, false);
    return d2[0];                              // = total, in every lane
}

__device__ __forceinline__ float block_reduce(float v, float* lds8) {
    const int lane = threadIdx.x & 31;
    const int wave = threadIdx.x >> 5;
    float ws = wave_reduce_wmma(v);
    if (lane == 0) lds8[wave] = ws;
    __syncthreads();
    float s = 0.0f;
    if (threadIdx.x == 0) {
        #pragma unroll
        for (int w = 0; w < THREADS / 32; ++w) s += lds8[w];
    }
    return s;                                  // valid in thread 0 only
}

__global__ __launch_bounds__(THREADS) void max_loss_main(
    const float* __restrict__ recon, const float* __restrict__ orig,
    float* __restrict__ partial) {
    const int gtid = blockIdx.x * THREADS + threadIdx.x;
    const int nthr = BLOCKS * THREADS;

    float acc = 0.0f;
    // Fixed trip count (uniform scalar loop, no exec masking): every thread
    // processes exactly ITERS chunks -> EXEC stays all-1s throughout.
    int c = gtid;
    #pragma unroll 2
    for (int it = 0; it < ITERS; ++it, c += nthr) {
        const int e  = c << 2;                 // element index
        const int y0 = e & (W_DIM - 1);        // chunks never straddle rows

        const float4 o4 = reinterpret_cast<const float4*>(orig)[c];
        const float4 r4 = reinterpret_cast<const float4*>(recon)[c];

        // Clamped-address neighbor loads + select (branch-free).
        const bool row_lo = (y0 == 0);
        const bool row_hi = (y0 == W_DIM - 4);
        float o_left  = orig[e - (row_lo ? 0 : 1)];
        float o_right = orig[e + (row_hi ? 3 : 4)];
        o_left  = row_lo ? 0.0f : o_left;      // zero-pad at row start
        o_right = row_hi ? 0.0f : o_right;     // zero-pad at row end

        const float o[4] = {o4.x, o4.y, o4.z, o4.w};
        const float r[4] = {r4.x, r4.y, r4.z, r4.w};

        // src: original where (nonzero && y>=1). Only element y0==0 is
        // affected by the y>=1 clause (y0 multiple of 4); o_left is already
        // zeroed at y0==0.
        float s[6];
        s[0] = o_left;
        s[1] = row_lo ? 0.0f : o[0];
        s[2] = o[1];  s[3] = o[2];  s[4] = o[3];
        s[5] = o_right;

        #pragma unroll
        for (int i = 0; i < 4; ++i) {
            // priority: src[y+1] > src[y] > src[y-1] > original[y]
            float om = o[i];
            om = (s[i]     != 0.0f) ? s[i]     : om;
            om = (s[i + 1] != 0.0f) ? s[i + 1] : om;
            om = (s[i + 2] != 0.0f) ? s[i + 2] : om;

            const float d0 = r[i] - o[i];
            const float m0 = d0 * d0;
            const float d1 = r[i] - om;
            const float m1 = d1 * d1 * d1 + CLOSE_MIN;
            float l = fminf(m0, m1);
            l = (o[i] != 0.0f) ? l * SIG_W : l;
            acc += l;
        }
    }

    __shared__ float lds8[THREADS / 32];
    const float bsum = block_reduce(acc, lds8);
    if (threadIdx.x == 0) partial[blockIdx.x] = bsum;
}

__global__ __launch_bounds__(THREADS) void max_loss_finish(
    const float* __restrict__ partial, float* __restrict__ out) {
    float acc = 0.0f;
    // Fixed trip count: uniform, EXEC all-1s at the WMMA.
    #pragma unroll
    for (int it = 0; it < FIN_ITERS; ++it)
        acc += partial[it * THREADS + threadIdx.x];

    __shared__ float lds8[THREADS / 32];
    const float total = block_reduce(acc, lds8);
    if (threadIdx.x == 0)
        out[0] = total * (1.0f / (float)N_ELEM);   // 2^-24, exact
}

extern "C" void kernel_launch(void* const* d_in, const int* in_sizes, int n_in,
                              void* d_out, int out_size, void* d_ws, size_t ws_size,
                              hipStream_t stream) {
    const float* recon = (const float*)d_in[0];
    const float* orig  = (const float*)d_in[1];
    float* out     = (float*)d_out;
    float* partial = (float*)d_ws;             // BLOCKS * 4 bytes = 8 KB

    max_loss_main<<<BLOCKS, THREADS, 0, stream>>>(recon, orig, partial);
    max_loss_finish<<<1, THREADS, 0, stream>>>(partial, out);
}